// EPGSimulation_12781822672902
// MI455X (gfx1250) — compile-verified
//
#include <hip/hip_runtime.h>

#define N_STEPS  8192
#define N_STATES 512
#define CHUNK    512

typedef unsigned int u32x4 __attribute__((ext_vector_type(4)));
typedef int          i32x4 __attribute__((ext_vector_type(4)));
typedef int          i32x8 __attribute__((ext_vector_type(8)));

// One workgroup, one lane per EPG order k. State (Fp,Fm,Z) lives in VGPRs;
// the +-1 k-shift goes through the LDS output staging buffer (which is laid
// out exactly as the [state][6] output tile), and each step's 12KB tile is
// streamed to HBM by the Tensor Data Mover (tensor_store_from_lds), double
// buffered and tracked with TENSORcnt so stores overlap the serial compute.
__global__ __launch_bounds__(N_STATES, 1)
void epg_scan_kernel(const float* __restrict__ flip,
                     const float* __restrict__ phase,
                     const float* __restrict__ pT1,
                     const float* __restrict__ pT2,
                     const float* __restrict__ pTR,
                     const float* __restrict__ pB0,
                     const float* __restrict__ pB1,
                     float* __restrict__ out)
{
    __shared__ float stage[2][N_STATES * 6];  // 24 KB: (Fp.re,Fp.im,Fm.re,Fm.im,Z.re,Z.im)
    __shared__ float scoef[CHUNK * 8];        // 16 KB: per-step RF coefficients

    const int k = (int)threadIdx.x;

    // Scalar setup (uniform).
    const float T1 = pT1[0], T2 = pT2[0], TR = pTR[0];
    const float B0 = pB0[0], B1 = pB1[0];
    const float E1 = expf(-TR / T1);
    const float E2 = expf(-TR / T2);
    const float one_m_E1 = 1.0f - E1;
    const float phi = 6.28318530717958647692f * B0 * TR * 0.001f;
    const float er  = E2 * cosf(phi);   // E2 * rot_p  (Fp gets +phi, Fm gets -phi)
    const float ei  = E2 * sinf(phi);

    // State registers.
    float Fre = 0.f, Fim = 0.f, Gre = 0.f, Gim = 0.f;
    float Zre = (k == 0) ? 1.f : 0.f, Zim = 0.f;

    const unsigned long long out_base = (unsigned long long)(uintptr_t)out;

    for (int n = 0; n < N_STEPS; ++n) {
        // ---- every CHUNK steps: precompute RF coefficients in parallel ----
        if ((n & (CHUNK - 1)) == 0) {
            __syncthreads();                      // prior readers of scoef done
            const int   t   = n + k;              // k < 512 == CHUNK
            const float ah  = flip[t] * B1 * 0.5f;
            const float ph  = phase[t];
            const float c   = cosf(ah), s  = sinf(ah);
            const float cb  = cosf(ph), sb = sinf(ph);
            float* cf = &scoef[k * 8];
            cf[0] = c * c;             // c2
            cf[1] = s * s;             // s2
            cf[2] = c * s;             // cs
            cf[3] = c * c - s * s;     // c2 - s2
            cf[4] = cb;                // cos(beta)
            cf[5] = sb;                // sin(beta)
            cf[6] = cb * cb - sb * sb; // cos(2beta)
            cf[7] = 2.f * cb * sb;     // sin(2beta)
            __syncthreads();
        }

        const float* cf = &scoef[(n & (CHUNK - 1)) * 8];
        const float c2 = cf[0], s2 = cf[1], cs = cf[2], c2ms2 = cf[3];
        const float cb = cf[4], sb = cf[5], c2b = cf[6], s2b = cf[7];

        // ---- relaxation + B0 precession ----
        const float fr = er * Fre - ei * Fim;
        const float fi = er * Fim + ei * Fre;
        const float gr = er * Gre + ei * Gim;   // conj rotation for Fm
        const float gi = er * Gim - ei * Gre;
        const float zr = fmaf(E1, Zre, one_m_E1);
        const float zi = E1 * Zim;

        // ---- RF mixing ----
        const float a1 = fmaf(gr, c2b,  gi * s2b);   // Re(conj(Fm)*e^{i2b})
        const float a2 = fmaf(gr, s2b, -gi * c2b);   // Im(conj(Fm)*e^{i2b})
        const float b1 = fmaf(fr, c2b, -fi * s2b);   // Re(conj(Fp)*e^{-i2b})
        const float b2 = fmaf(fr, s2b,  fi * c2b);   // -Im(conj(Fp)*e^{-i2b})
        const float p1 = zr * cb, p2 = zi * sb, p3 = zr * sb, p4 = zi * cb;
        const float zc1 = p1 - p2, zc2 = p3 + p4;    // Z*e^{ib}
        const float zd1 = p1 + p2, zd2 = p4 - p3;    // Z*e^{-ib}
        const float f1 = fmaf(fr, cb,  fi * sb);     // Fp*e^{-ib}
        const float f2 = fmaf(fi, cb, -fr * sb);
        const float g1 = fmaf(gr, cb, -gi * sb);     // Fm*e^{ib}
        const float g2 = fmaf(gr, sb,  gi * cb);

        const float Fpn_re = fmaf(c2, fr, fmaf(s2, a1, -cs * zc2));
        const float Fpn_im = fmaf(c2, fi, fmaf(s2, a2,  cs * zc1));
        const float Fmn_re = fmaf(s2, b1, fmaf(c2, gr,  cs * zd2));
        const float Fmn_im = fmaf(-s2, b2, fmaf(c2, gi, -cs * zd1));
        const float Zn_re  = fmaf(cs,  f2 - g2, c2ms2 * zr);
        const float Zn_im  = fmaf(-cs, f1 - g1, c2ms2 * zi);

        // ---- gate: TDM store that read this parity's buffer (2 steps ago)
        // must have drained before we overwrite it ----
        if ((threadIdx.x >> 5) == 0) {
            __builtin_amdgcn_s_wait_tensorcnt(1);
        }
        __syncthreads();

        // ---- shifted write into staging (already in output layout) ----
        float* st = stage[n & 1];
        if (k == N_STATES - 1) { st[0] = 0.f; st[1] = 0.f; }              // Fp[0]=0
        else { st[(k + 1) * 6 + 0] = Fpn_re; st[(k + 1) * 6 + 1] = Fpn_im; }
        if (k == 0) { st[(N_STATES - 1) * 6 + 2] = 0.f;                    // Fm[last]=0
                      st[(N_STATES - 1) * 6 + 3] = 0.f; }
        else { st[(k - 1) * 6 + 2] = Fmn_re; st[(k - 1) * 6 + 3] = Fmn_im; }
        st[k * 6 + 4] = Zn_re;
        st[k * 6 + 5] = Zn_im;
        __syncthreads();

        // ---- read back shifted state for the next step ----
        Fre = st[k * 6 + 0]; Fim = st[k * 6 + 1];
        Gre = st[k * 6 + 2]; Gim = st[k * 6 + 3];
        Zre = Zn_re;         Zim = Zn_im;

        // ---- wave 0 fires the async TDM store of this step's 12KB tile ----
        if ((threadIdx.x >> 5) == 0) {
            const unsigned long long gaddr =
                out_base + (unsigned long long)n * (N_STATES * 6u * 4u);
            const unsigned lds_off = (unsigned)(uintptr_t)st;
            // D# group0: count=1, lds_addr, global_addr[56:0], type=2
            u32x4 g0 = { 1u,
                         lds_off,
                         (unsigned)gaddr,
                         (unsigned)((gaddr >> 32) & 0x1FFFFFFull) | 0x80000000u };
            // D# group1: data_size=4B; tensor_dim0 = tile_dim0 = 3072 dwords;
            // tensor_dim1 = tile_dim1 = 1 row; dim0_stride = 3072.
            i32x8 g1 = { 0x00020000,
                         (int)(3072u << 16),   // tensor_dim0 low16 in [31:16]
                         0x00010000,           // tensor_dim1 = 1
                         (int)(3072u << 16),   // tile_dim0
                         1,                    // tile_dim1 = 1
                         3072,                 // tensor_dim0_stride
                         0, 0 };
            i32x4 gz4 = { 0, 0, 0, 0 };
            i32x8 gz8 = { 0, 0, 0, 0, 0, 0, 0, 0 };
            // clang-23 / therock-10.0 form: 6 args
            __builtin_amdgcn_tensor_store_from_lds(g0, g1, gz4, gz4, gz8, 0);
        }
    }

    if ((threadIdx.x >> 5) == 0) {
        __builtin_amdgcn_s_wait_tensorcnt(0);   // drain before endpgm
    }
}

extern "C" void kernel_launch(void* const* d_in, const int* in_sizes, int n_in,
                              void* d_out, int out_size, void* d_ws, size_t ws_size,
                              hipStream_t stream) {
    (void)in_sizes; (void)n_in; (void)out_size; (void)d_ws; (void)ws_size;
    // setup_inputs order: flip_angles, phases, T1, T2, TR, TE(unused), B0, B1
    epg_scan_kernel<<<dim3(1), dim3(N_STATES), 0, stream>>>(
        (const float*)d_in[0], (const float*)d_in[1],
        (const float*)d_in[2], (const float*)d_in[3],
        (const float*)d_in[4], (const float*)d_in[6],
        (const float*)d_in[7], (float*)d_out);
}